// NeuralODE_42279658061875
// MI455X (gfx1250) — compile-verified
//
#include <hip/hip_runtime.h>
#include <math.h>

typedef __attribute__((ext_vector_type(2))) float v2f;
typedef __attribute__((ext_vector_type(8))) float v8f;

#define BATCH 32
#define NDIM 256
#define LDS_STRIDE 258   // padded row stride (floats) to avoid 16-way bank conflicts

#if __has_builtin(__builtin_amdgcn_tanhf)
#define TANHF(x) __builtin_amdgcn_tanhf(x)
#else
#define TANHF(x) tanhf(x)
#endif

struct StageCoefs {
  float c[5];     // dt * a_ij for j < ncoef
  int   ncoef;    // stage index i (number of prior k's)
  float bscale;   // scalar c_i * R^n applied to bias AND adj at this (step, stage)
};

// One block per l (256 blocks). Computes k_i[l, b] for all b:
//   z[b,m]  = tanh( sum_j s_i[b,j] * K[l*256+m, j] )       (fp32 WMMA 16x16x4)
//   k_i     = bscale * ( bias[b,l] - sum_m adj[b,l,m] * z[b,m] )
__global__ __launch_bounds__(256) void node_stage_kernel(
    const float* __restrict__ Kw,     // [N*N, N]  kernel[l,m,j] flattened
    const float* __restrict__ bias,   // [B, N]
    const float* __restrict__ adj,    // [B, N, N]
    const float* __restrict__ s,      // [B, N] base state of this step
    const float* __restrict__ kbase,  // k buffers: 6 x [N][B] (l-major, b-minor)
    float* __restrict__ kout,         // [N][B]
    StageCoefs sc) {
  __shared__ float sLDS[BATCH * LDS_STRIDE];
  __shared__ float dsacc[BATCH];

  const int tid = threadIdx.x;
  const int l = blockIdx.x;

  // Build stage state s_i into LDS (rows = b, padded stride).
  for (int e = tid; e < BATCH * NDIM; e += 256) {
    const int b = e >> 8;
    const int j = e & 255;
    float v = s[e];
    for (int q = 0; q < sc.ncoef; ++q)
      v += sc.c[q] * kbase[q * (BATCH * NDIM) + j * BATCH + b];
    sLDS[b * LDS_STRIDE + j] = v;
  }
  if (tid < BATCH) dsacc[tid] = 0.0f;
  __syncthreads();

  const int lane = tid & 31;
  const int w = tid >> 5;            // wave 0..7
  const int half = lane >> 4;        // 0 or 1 (lane half)
  const int r = lane & 15;
  const int m0 = (2 * w) * 16 + r;   // lane's column in first m-tile
  const int m1 = m0 + 16;            // second m-tile

  // B operand: K rows p = l*256 + m, columns j + 2*half .. +1  (16x16x4 layout)
  const float* kr0 = Kw + ((size_t)(l * NDIM + m0)) * NDIM + 2 * half;
  const float* kr1 = Kw + ((size_t)(l * NDIM + m1)) * NDIM + 2 * half;
  // A operand from LDS: row r (b-tile 0) and row 16+r (b-tile 1)
  const float* a0p = &sLDS[r * LDS_STRIDE + 2 * half];
  const float* a1p = &sLDS[(16 + r) * LDS_STRIDE + 2 * half];

  v8f acc00, acc01, acc10, acc11;
  for (int g = 0; g < 8; ++g) { acc00[g] = 0.f; acc01[g] = 0.f; acc10[g] = 0.f; acc11[g] = 0.f; }

#pragma unroll 4
  for (int j = 0; j < NDIM; j += 4) {
    v2f a0 = *(const v2f*)(a0p + j);
    v2f a1 = *(const v2f*)(a1p + j);
    v2f b0 = *(const v2f*)(kr0 + j);
    v2f b1 = *(const v2f*)(kr1 + j);
    acc00 = __builtin_amdgcn_wmma_f32_16x16x4_f32(false, a0, false, b0, (short)0, acc00, false, false);
    acc01 = __builtin_amdgcn_wmma_f32_16x16x4_f32(false, a0, false, b1, (short)0, acc01, false, false);
    acc10 = __builtin_amdgcn_wmma_f32_16x16x4_f32(false, a1, false, b0, (short)0, acc10, false, false);
    acc11 = __builtin_amdgcn_wmma_f32_16x16x4_f32(false, a1, false, b1, (short)0, acc11, false, false);
  }

  // Fused epilogue: tanh, multiply by adj, reduce over m.
  const size_t adjL = (size_t)l * NDIM;
#pragma unroll
  for (int g = 0; g < 8; ++g) {
    const int b0r = g + 8 * half;        // rows 0..15 of C/D tile (b-tile 0)
    const int b1r = 16 + g + 8 * half;   // b-tile 1
    float p0 = TANHF(acc00[g]) * adj[(size_t)b0r * (NDIM * NDIM) + adjL + m0]
             + TANHF(acc01[g]) * adj[(size_t)b0r * (NDIM * NDIM) + adjL + m1];
    float p1 = TANHF(acc10[g]) * adj[(size_t)b1r * (NDIM * NDIM) + adjL + m0]
             + TANHF(acc11[g]) * adj[(size_t)b1r * (NDIM * NDIM) + adjL + m1];
#pragma unroll
    for (int mask = 1; mask < 16; mask <<= 1) {
      p0 += __shfl_xor(p0, mask, 32);
      p1 += __shfl_xor(p1, mask, 32);
    }
    if (r == 0) {
      atomicAdd(&dsacc[b0r], p0);   // ds_add_f32
      atomicAdd(&dsacc[b1r], p1);
    }
  }
  __syncthreads();

  if (tid < BATCH)
    kout[l * BATCH + tid] = sc.bscale * (bias[tid * NDIM + l] - dsacc[tid]);
}

// Per-step state update: s += dt * sum(bw_i * k_i); optionally write save row.
__global__ __launch_bounds__(256) void node_update_kernel(
    float* __restrict__ s, const float* __restrict__ kbase, float* __restrict__ save) {
  const int e = blockIdx.x * 256 + threadIdx.x;   // 8192 threads
  const int b = e >> 8;
  const int j = e & 255;
  const int idx = j * BATCH + b;
  const int KS = BATCH * NDIM;
  float acc = (float)(35.0 / 384.0)    * kbase[idx]
            + (float)(500.0 / 1113.0)  * kbase[2 * KS + idx]
            + (float)(125.0 / 192.0)   * kbase[3 * KS + idx]
            + (float)(-2187.0 / 6784.0)* kbase[4 * KS + idx]
            + (float)(11.0 / 84.0)     * kbase[5 * KS + idx];
  const float v = s[e] + 0.01f * acc;
  s[e] = v;
  if (save) save[e] = v;
}

__global__ __launch_bounds__(256) void node_init_kernel(
    const float* __restrict__ y0, float* __restrict__ s, float* __restrict__ out0) {
  const int e = blockIdx.x * 256 + threadIdx.x;
  const float v = y0[e];
  s[e] = v;
  out0[e] = v;
}

extern "C" void kernel_launch(void* const* d_in, const int* in_sizes, int n_in,
                              void* d_out, int out_size, void* d_ws, size_t ws_size,
                              hipStream_t stream) {
  // inputs: ts (unused), y0 [B,N], bias [B,N,1], data_adj [B,N,N], kernel [N,N,N]
  const float* y0   = (const float*)d_in[1];
  const float* bias = (const float*)d_in[2];
  const float* adj  = (const float*)d_in[3];
  const float* Kw   = (const float*)d_in[4];
  float* out = (float*)d_out;

  float* s     = (float*)d_ws;            // [B*N]
  float* kbase = s + BATCH * NDIM;        // 6 x [N*B]   (total ws use: 224 KB)

  node_init_kernel<<<(BATCH * NDIM) / 256, 256, 0, stream>>>(y0, s, out);

  static const double A[6][5] = {
      {},
      {1.0 / 5.0},
      {3.0 / 40.0, 9.0 / 40.0},
      {44.0 / 45.0, -56.0 / 15.0, 32.0 / 9.0},
      {19372.0 / 6561.0, -25360.0 / 2187.0, 64448.0 / 6561.0, -212.0 / 729.0},
      {9017.0 / 3168.0, -355.0 / 33.0, 46732.0 / 5247.0, 49.0 / 176.0, -5103.0 / 18656.0}};
  static const double BW[6] = {35.0 / 384.0, 0.0, 500.0 / 1113.0,
                               125.0 / 192.0, -2187.0 / 6784.0, 11.0 / 84.0};
  const double dt = 0.01;

  // bias/adj obey ydot = y: stage value = y0 * C[i] * R^n (pure scalars).
  double C[6];
  C[0] = 1.0;
  for (int i = 1; i < 6; ++i) {
    double acc = 0.0;
    for (int q = 0; q < i; ++q) acc += A[i][q] * C[q];
    C[i] = 1.0 + dt * acc;
  }
  double R = 0.0;
  for (int i = 0; i < 6; ++i) R += BW[i] * C[i];
  R = 1.0 + dt * R;

  double Rn = 1.0;
  int save = 0;
  for (int n = 0; n < 100; ++n) {
    for (int i = 0; i < 6; ++i) {
      StageCoefs sc;
      for (int q = 0; q < 5; ++q) sc.c[q] = (q < i) ? (float)(dt * A[i][q]) : 0.0f;
      sc.ncoef = i;
      sc.bscale = (float)(C[i] * Rn);
      node_stage_kernel<<<NDIM, 256, 0, stream>>>(
          Kw, bias, adj, s, kbase, kbase + i * BATCH * NDIM, sc);
    }
    Rn *= R;
    float* savep = nullptr;
    if ((n + 1) % 10 == 0) {
      ++save;
      savep = out + (size_t)save * BATCH * NDIM;
    }
    node_update_kernel<<<BATCH, 256, 0, stream>>>(s, kbase, savep);
  }
}